// GCNN_44744969290030
// MI455X (gfx1250) — compile-verified
//
#include <hip/hip_runtime.h>

// Problem constants (match reference setup_inputs)
#define N_NODES 50000
#define N_EDGES 800000
#define NB      100000u   // n*B = rows of the permuted GEMM input M
#define C_OUT   64

typedef __attribute__((ext_vector_type(2))) float v2f;
typedef __attribute__((ext_vector_type(8))) float v8f;

// ---------------------------------------------------------------------------
// Kernel 0: zero the L2-resident accumulator (d_ws is not pre-zeroed).
// ---------------------------------------------------------------------------
__global__ void GCNN_zero_kernel(float* __restrict__ p, int n) {
    int i = blockIdx.x * blockDim.x + threadIdx.x;
    if (i < n) p[i] = 0.0f;
}

// ---------------------------------------------------------------------------
// Kernel 1: COO SpMM scatter. agg[i, k] += val_e * h[j, k],
// h[j, k] = x[k>>6, j, k&63]. One thread per (edge, channel):
// 800k * 128 = 102.4M f32 atomics; agg (25.6 MB) lives in the 192 MB L2,
// so the RMW traffic never leaves the chip.
// ---------------------------------------------------------------------------
__global__ void GCNN_spmm_scatter_kernel(const float* __restrict__ x,
                                         const int*   __restrict__ rows,
                                         const int*   __restrict__ cols,
                                         const float* __restrict__ vals,
                                         float*       __restrict__ agg) {
    long long g = (long long)blockIdx.x * blockDim.x + threadIdx.x;
    int e = (int)(g >> 7);      // edge index
    int k = (int)(g & 127);     // channel 0..127  (k = b*64 + c)
    if (e >= N_EDGES) return;
    int   i = rows[e];
    int   j = cols[e];
    float v = vals[e];
    float hv = x[((k >> 6) * N_NODES + j) * 64 + (k & 63)];
    atomicAdd(&agg[i * 128 + k], v * hv);
}

// ---------------------------------------------------------------------------
// Kernel 2: fused permutation-gather + GEMM (fp32 WMMA) + bias + ReLU.
//
// The reference's reshape/transpose bookkeeping reduces to:
//   M[r, c] = agg[i*128 + 2*c0 + (f & 1)]
//   with f = r*64 + c, c0 = f / 100000, i = (f % 100000) >> 1.
// out[r, co] = relu(sum_c M[r,c] * W[c,co] + bias[co]),
// and d_out flat layout == [100000, 64] row-major == [B, n, C_out].
//
// One wave per 16x16 output tile, V_WMMA_F32_16X16X4_F32 over K=64 (16 MMAs).
// Block = 128 threads = 4 waves covering all 64 output columns of 16 M-rows.
// ---------------------------------------------------------------------------
__global__ void GCNN_gemm_wmma_kernel(const float* __restrict__ agg,
                                      const float* __restrict__ W,
                                      const float* __restrict__ bias,
                                      float*       __restrict__ out) {
    const int lane = threadIdx.x & 31;
    const int wave = threadIdx.x >> 5;
    const int half = lane >> 4;      // 0: K pair {k0,k0+1}, 1: {k0+2,k0+3}
    const int l16  = lane & 15;
    const unsigned r0 = (unsigned)blockIdx.x * 16u;   // M-tile base row
    const int      n0 = wave * 16;                    // N-tile base col
    const unsigned r  = r0 + (unsigned)l16;           // this lane's A row

    v8f acc = {};
    #pragma unroll
    for (int k0 = 0; k0 < 64; k0 += 4) {
        const int ka = k0 + 2 * half;
        v2f a, b;
        // A operand: gathered M[r, ka], M[r, ka+1]
        {
            unsigned f  = r * 64u + (unsigned)ka;
            unsigned c0 = f / NB;
            unsigned rem = f - c0 * NB;
            a.x = agg[(rem >> 1) * 128u + c0 * 2u + (f & 1u)];
            f += 1u;
            c0  = f / NB;
            rem = f - c0 * NB;
            a.y = agg[(rem >> 1) * 128u + c0 * 2u + (f & 1u)];
        }
        // B operand: W[ka, n0+l16], W[ka+1, n0+l16]
        b.x = W[ka * C_OUT + n0 + l16];
        b.y = W[(ka + 1) * C_OUT + n0 + l16];

        // D = A(16x4) * B(4x16) + C, full fp32
        acc = __builtin_amdgcn_wmma_f32_16x16x4_f32(
            /*neg_a=*/false, a, /*neg_b=*/false, b,
            /*c_mod=*/(short)0, acc, /*reuse_a=*/false, /*reuse_b=*/false);
    }

    // Epilogue: C/D layout -> lane<16: (M=v, N=lane); lane>=16: (M=v+8, N=lane-16)
    const int   col = n0 + l16;
    const float bv  = bias[col];
    #pragma unroll
    for (int v = 0; v < 8; ++v) {
        float o = acc[v] + bv;
        o = o > 0.0f ? o : 0.0f;
        out[(r0 + (unsigned)(v + 8 * half)) * (unsigned)C_OUT + (unsigned)col] = o;
    }
}

// ---------------------------------------------------------------------------
// Host launcher. Inputs (setup_inputs order):
//   0: x [2,50000,64] f32   1: adj_rows [800k] i32   2: adj_cols [800k] i32
//   3: adj_vals [800k] f32  4: weights [64,64] f32   5: bias [64] f32
// Output: [2,50000,64] f32. Workspace: agg [50000,128] f32 = 25.6 MB.
// ---------------------------------------------------------------------------
extern "C" void kernel_launch(void* const* d_in, const int* in_sizes, int n_in,
                              void* d_out, int out_size, void* d_ws, size_t ws_size,
                              hipStream_t stream) {
    const float* x    = (const float*)d_in[0];
    const int*   rows = (const int*)  d_in[1];
    const int*   cols = (const int*)  d_in[2];
    const float* vals = (const float*)d_in[3];
    const float* W    = (const float*)d_in[4];
    const float* bias = (const float*)d_in[5];
    float* out = (float*)d_out;
    float* agg = (float*)d_ws;      // 50000*128 floats

    const int aggN = N_NODES * 128;
    GCNN_zero_kernel<<<(aggN + 255) / 256, 256, 0, stream>>>(agg, aggN);

    const long long scatterThreads = (long long)N_EDGES * 128;
    const int scatterBlocks = (int)((scatterThreads + 255) / 256);
    GCNN_spmm_scatter_kernel<<<scatterBlocks, 256, 0, stream>>>(x, rows, cols,
                                                               vals, agg);

    GCNN_gemm_wmma_kernel<<<NB / 16, 128, 0, stream>>>(agg, W, bias, out);
}